// HitAvailabilityOneHot_10582799418093
// MI455X (gfx1250) — compile-verified
//
#include <hip/hip_runtime.h>

#define NCOMB 21
#define BOARD 27
#define BLK   256

static constexpr int DW_PER_BLOCK = BLK * BOARD;      // 6912 dwords = 27648 B of LDS
static constexpr int V4_PER_BLOCK = DW_PER_BLOCK / 4; // 1728 x b128 async copies

typedef int v4i __attribute__((vector_size(16)));           // int4, as the builtin wants
typedef __attribute__((address_space(3))) v4i as3_v4i;      // LDS-side pointee
typedef __attribute__((address_space(3))) char as3_char;

// ---------------------------------------------------------------------------
// Kernel 1: per-combo min of deltas[:, :, 0] and max of deltas[:, :, 25]
// 21 blocks x 256 threads, LDS tree reduction. Tiny (<1% of runtime).
// ---------------------------------------------------------------------------
__global__ void __launch_bounds__(BLK) dice_reduce_kernel(
    const float* __restrict__ deltas, float* __restrict__ ws, int K) {
  __shared__ float smin[BLK];
  __shared__ float smax[BLK];
  const int n = blockIdx.x;   // 0..20
  const int t = threadIdx.x;
  const float* base = deltas + (size_t)n * (size_t)K * BOARD;
  float mn =  3.402823466e+38f;
  float mx = -3.402823466e+38f;
  for (int k = t; k < K; k += BLK) {
    mn = fminf(mn, base[(size_t)k * BOARD + 0]);
    mx = fmaxf(mx, base[(size_t)k * BOARD + 25]);
  }
  smin[t] = mn; smax[t] = mx;
  __syncthreads();
  for (int s = BLK / 2; s > 0; s >>= 1) {
    if (t < s) {
      smin[t] = fminf(smin[t], smin[t + s]);
      smax[t] = fmaxf(smax[t], smax[t + s]);
    }
    __syncthreads();
  }
  if (t == 0) {
    ws[n]         = smin[0];  // min_d0[n]
    ws[NCOMB + n] = smax[0];  // max_d25[n]
  }
}

// ---------------------------------------------------------------------------
// Kernel 2: bandwidth-bound main pass.
// Async-copy (GLOBAL_LOAD_ASYNC_TO_LDS_B128, ASYNCcnt) each block's 256x27
// float tile of x into LDS fully coalesced, then each thread evaluates the
// 21 dice combos for its row and stores a float4.
// ---------------------------------------------------------------------------
__global__ void __launch_bounds__(BLK) dice_onehot_kernel(
    const float* __restrict__ x, const float* __restrict__ ws,
    float4* __restrict__ out, int M) {
  __shared__ float lds[DW_PER_BLOCK];
  const int tid = threadIdx.x;
  const long long blockRow = (long long)blockIdx.x * BLK;
  const long long rem = (long long)M - blockRow;
  const int rows = rem < BLK ? (int)rem : BLK;
  const float* src = x + blockRow * BOARD;

  float x0 = 0.0f, x25 = 0.0f;
  if (rows == BLK) {
    as3_char* lbase = (as3_char*)lds;
    const char* g = (const char*)src;
    for (int i = tid; i < V4_PER_BLOCK; i += BLK) {
#if __has_builtin(__builtin_amdgcn_global_load_async_to_lds_b128)
      __builtin_amdgcn_global_load_async_to_lds_b128(
          (v4i*)(g + (size_t)i * 16),
          (as3_v4i*)(lbase + i * 16),
          /*offset=*/0, /*cpol=*/0);
#else
      unsigned loff = (unsigned)(size_t)(lbase + i * 16);
      unsigned long long ga = (unsigned long long)g + (unsigned long long)i * 16;
      asm volatile("global_load_async_to_lds_b128 %0, %1, off"
                   :: "v"(loff), "v"(ga) : "memory");
#endif
    }
#if __has_builtin(__builtin_amdgcn_s_wait_asynccnt)
    __builtin_amdgcn_s_wait_asynccnt(0);
#else
    asm volatile("s_wait_asynccnt 0" ::: "memory");
#endif
    __syncthreads();
    // stride-27 dword reads: gcd(27,64)=1 -> bank-conflict free in wave32
    x0  = lds[tid * BOARD + 0];
    x25 = lds[tid * BOARD + 25];
  } else if (tid < rows) {
    // tail block (not hit when M % 256 == 0): plain loads, no OOB b128 risk
    x0  = src[(size_t)tid * BOARD + 0];
    x25 = src[(size_t)tid * BOARD + 25];
  }

  if (tid < rows) {
    // dice factors: (d1==d2 ? 1 : 2)/36, d1<=d2 in row-major pair order
    const float F[NCOMB] = {
      1.f/36, 2.f/36, 2.f/36, 2.f/36, 2.f/36, 2.f/36,
      1.f/36, 2.f/36, 2.f/36, 2.f/36, 2.f/36,
      1.f/36, 2.f/36, 2.f/36, 2.f/36,
      1.f/36, 2.f/36, 2.f/36,
      1.f/36, 2.f/36,
      1.f/36 };
    float s0 = 0.f, s1 = 0.f, s2 = 0.f, s3 = 0.f;
#pragma unroll
    for (int n = 0; n < NCOMB; ++n) {
      const float a = x0  + ws[n];          // uniform -> scalar loads
      const float b = x25 + ws[NCOMB + n];
      s0 += (a <  -1.0f) ? F[n] : 0.0f;
      s1 += (a == -1.0f) ? F[n] : 0.0f;
      s2 += (b ==  1.0f) ? F[n] : 0.0f;
      s3 += (b >   1.0f) ? F[n] : 0.0f;
    }
    out[blockRow + tid] = make_float4(s0, s1, s2, s3);
  }
}

extern "C" void kernel_launch(void* const* d_in, const int* in_sizes, int n_in,
                              void* d_out, int out_size, void* d_ws, size_t ws_size,
                              hipStream_t stream) {
  (void)n_in; (void)out_size; (void)ws_size;
  const float* x      = (const float*)d_in[0];   // (M, 27) f32
  const float* deltas = (const float*)d_in[1];   // (21, K, 27) f32
  float* ws = (float*)d_ws;                      // 42 floats of scratch

  const int M = in_sizes[0] / BOARD;
  const int K = in_sizes[1] / (NCOMB * BOARD);

  dice_reduce_kernel<<<NCOMB, BLK, 0, stream>>>(deltas, ws, K);

  const int grid = (M + BLK - 1) / BLK;
  dice_onehot_kernel<<<grid, BLK, 0, stream>>>(x, ws, (float4*)d_out, M);
}